// GCNEncoder_74964359184927
// MI455X (gfx1250) — compile-verified
//
#include <hip/hip_runtime.h>
#include <hip/hip_bf16.h>

typedef float v2f __attribute__((ext_vector_type(2)));
typedef float v8f __attribute__((ext_vector_type(8)));

#define A_STRIDE 132  // A-tile row stride in dwords: (4r)%64 bank pattern -> conflict-free ds_load_b64
#define TILE_F   (16 * A_STRIDE)   // floats per A tile buffer (2112)

// ---------------------------------------------------------------- utility
__global__ void zero_f32(float* __restrict__ p, int n) {
    for (int i = blockIdx.x * blockDim.x + threadIdx.x; i < n; i += gridDim.x * blockDim.x)
        p[i] = 0.0f;
}

// ---------------------------------------------------------------- column stats (mean / unbiased var)
__global__ __launch_bounds__(128) void col_stats(const float* __restrict__ x,
                                                 float* __restrict__ sums,
                                                 float* __restrict__ sumsq, int N) {
    int f = threadIdx.x;
    float s = 0.0f, s2 = 0.0f;
    for (int i = blockIdx.x; i < N; i += gridDim.x) {
        float v = x[(size_t)i * 128 + f];
        s += v; s2 += v * v;
    }
    unsafeAtomicAdd(&sums[f], s);
    unsafeAtomicAdd(&sumsq[f], s2);
}

__global__ __launch_bounds__(128) void stats_finalize(const float* __restrict__ sums,
                                                      const float* __restrict__ sumsq,
                                                      float* __restrict__ inv_sigma,
                                                      float* __restrict__ mu_os, int N) {
    int f = threadIdx.x;
    float s = sums[f];
    float mu = s / (float)N;
    float var = (sumsq[f] - s * mu) / (float)(N - 1);  // ddof=1
    float inv = rsqrtf(var);
    inv_sigma[f] = inv;
    mu_os[f] = mu * inv;
}

__global__ __launch_bounds__(128) void build_w1p(const float* __restrict__ W1,
                                                 const float* __restrict__ inv_sigma,
                                                 float* __restrict__ W1p) {
    int k = blockIdx.x, n = threadIdx.x;
    W1p[k * 128 + n] = W1[k * 128 + n] * inv_sigma[k];
}

__global__ __launch_bounds__(128) void build_c1(const float* __restrict__ W1,
                                                const float* __restrict__ mu_os,
                                                float* __restrict__ c1) {
    int n = threadIdx.x;
    float acc = 0.0f;
    for (int k = 0; k < 128; ++k) acc += mu_os[k] * W1[k * 128 + n];
    c1[n] = -acc;
}

// ---------------------------------------------------------------- degree / normalization
__global__ void deg_count(const int* __restrict__ dst, float* __restrict__ deg, int E) {
    for (int e = blockIdx.x * blockDim.x + threadIdx.x; e < E; e += gridDim.x * blockDim.x)
        unsafeAtomicAdd(&deg[dst[e]], 1.0f);
}

__global__ void make_dinv(float* __restrict__ deg, int N) {
    for (int i = blockIdx.x * blockDim.x + threadIdx.x; i < N; i += gridDim.x * blockDim.x)
        deg[i] = rsqrtf(deg[i] + 1.0f);
}

__global__ void make_coef(const int* __restrict__ src, const int* __restrict__ dst,
                          const float* __restrict__ dinv, float* __restrict__ coef, int E) {
    for (int e = blockIdx.x * blockDim.x + threadIdx.x; e < E; e += gridDim.x * blockDim.x)
        coef[e] = dinv[src[e]] * dinv[dst[e]];
}

// ---------------------------------------------------------------- WMMA GEMM pieces
// Async stage of one 16x128 A tile into LDS (row stride A_STRIDE), 16B chunks, b128 async copies.
__device__ __forceinline__ void stage_async(const float* __restrict__ A, int row0, int N,
                                            unsigned ldsbuf, int tid) {
#pragma unroll
    for (int j = 0; j < 4; ++j) {
        int chunk = tid + 128 * j;           // 0..511 : 16 rows x 32 chunks (16B each)
        int r = chunk >> 5;
        int cc = chunk & 31;
        int row = row0 + r; if (row >= N) row = N - 1;   // clamped dummy rows: results discarded at store
        const float* gp = A + (size_t)row * 128 + cc * 4;
        unsigned lp = ldsbuf + (unsigned)(r * (A_STRIDE * 4) + cc * 16);
        asm volatile("global_load_async_to_lds_b128 %0, %1, off"
                     :: "v"(lp), "v"(gp) : "memory");
    }
}

__device__ __forceinline__ void tile_compute_store(const float* __restrict__ Atile,
                                                   const v2f (&bf0)[32], const v2f (&bf1)[32],
                                                   const float* __restrict__ storeAdd,
                                                   float* __restrict__ out, int row0, int N,
                                                   int lmod, int halfsel, int n0, int n1) {
    v8f acc0 = {}, acc1 = {};
#pragma unroll
    for (int kt = 0; kt < 32; ++kt) {
        int k0 = kt * 4 + 2 * halfsel;
        v2f a;                                  // A 16x4 frag: row=lmod, K=k0,k0+1 -> ds_load_b64
        a.x = Atile[lmod * A_STRIDE + k0];
        a.y = Atile[lmod * A_STRIDE + k0 + 1];
        acc0 = __builtin_amdgcn_wmma_f32_16x16x4_f32(false, a, false, bf0[kt], (short)0, acc0, false, false);
        acc1 = __builtin_amdgcn_wmma_f32_16x16x4_f32(false, a, false, bf1[kt], (short)0, acc1, false, false);
    }
    float add0 = storeAdd ? storeAdd[n0] : 0.0f;
    float add1 = storeAdd ? storeAdd[n1] : 0.0f;
#pragma unroll
    for (int r = 0; r < 8; ++r) {
        int row = row0 + r + 8 * halfsel;       // C/D layout: VGPR r -> row r + 8*halfsel
        if (row < N) {
            out[(size_t)row * 128 + n0] = acc0[r] + add0;
            out[(size_t)row * 128 + n1] = acc1[r] + add1;
        }
    }
}

// out[N,128] = f(A)[N,128] @ W[128,128] (+ storeAdd[n]).  ASYNC path: raw A via
// double-buffered global_load_async_to_lds_b128.  Sync path: f(A)=relu(A+reluBias).
template <bool ASYNC>
__global__ __launch_bounds__(128) void gemm128_wmma(const float* __restrict__ A,
                                                    const float* __restrict__ W,
                                                    const float* __restrict__ reluBias,
                                                    const float* __restrict__ storeAdd,
                                                    float* __restrict__ out, int N) {
    __shared__ float smem[2 * TILE_F];          // two A-tile buffers (~17KB)
    const int tid     = threadIdx.x;
    const int lane    = tid & 31;
    const int wave    = tid >> 5;               // 0..3, owns column tiles 2*wave, 2*wave+1
    const int halfsel = lane >> 4;
    const int lmod    = lane & 15;
    const int n0      = wave * 32 + lmod;
    const int n1      = n0 + 16;

    // B panels (128x16 each) straight from L2-cached global into registers.
    // B 4x16 frag layout: v.x = W[k0 + 2*halfsel][n], v.y = W[k0+1 + 2*halfsel][n]
    v2f bf0[32], bf1[32];
#pragma unroll
    for (int kt = 0; kt < 32; ++kt) {
        int k0 = kt * 4 + 2 * halfsel;
        bf0[kt].x = W[k0 * 128 + n0];
        bf0[kt].y = W[(k0 + 1) * 128 + n0];
        bf1[kt].x = W[k0 * 128 + n1];
        bf1[kt].y = W[(k0 + 1) * 128 + n1];
    }

    const int ntiles = (N + 15) >> 4;
    const unsigned ldsbase = (unsigned)(uintptr_t)(&smem[0]);

    if (ASYNC) {
        int t = blockIdx.x;
        if (t < ntiles) stage_async(A, t << 4, N, ldsbase, tid);
        int buf = 0;
        for (; t < ntiles; t += gridDim.x) {
            int tn = t + gridDim.x; if (tn >= ntiles) tn = t;      // dummy prefetch keeps counter math uniform
            stage_async(A, tn << 4, N, ldsbase + (unsigned)((buf ^ 1) * TILE_F * 4), tid);
            asm volatile("s_wait_asynccnt 0x4" ::: "memory");      // tile t's 4 copies retired
            __syncthreads();
            tile_compute_store(&smem[buf * TILE_F], bf0, bf1, storeAdd, out,
                               t << 4, N, lmod, halfsel, n0, n1);
            __syncthreads();                                       // done reading buf before overwrite
            buf ^= 1;
        }
    } else {
        for (int t = blockIdx.x; t < ntiles; t += gridDim.x) {
            const int row0 = t << 4;
            for (int i = tid; i < 2048; i += 128) {                // 16 elems/thread, coalesced
                int r = i >> 7, c = i & 127;
                int row = row0 + r;
                float v = 0.0f;
                if (row < N) {
                    v = A[(size_t)row * 128 + c];
                    if (reluBias) { v += reluBias[c]; v = v > 0.0f ? v : 0.0f; }
                }
                smem[r * A_STRIDE + c] = v;
            }
            __syncthreads();
            tile_compute_store(smem, bf0, bf1, storeAdd, out, row0, N, lmod, halfsel, n0, n1);
            __syncthreads();
        }
    }
}

// ---------------------------------------------------------------- aggregation
// agg[i][f] = xw[i][f] * dinv[i]^2   (self-loop term doubles as zero-init); float4 lanes
__global__ void self_init(const float4* __restrict__ xw, const float* __restrict__ dinv,
                          float4* __restrict__ agg, int total /* N*32 */) {
    for (int g = blockIdx.x * blockDim.x + threadIdx.x; g < total; g += gridDim.x * blockDim.x) {
        int i = g >> 5;
        float d = dinv[i]; d *= d;
        float4 v = xw[g];
        v.x *= d; v.y *= d; v.z *= d; v.w *= d;
        agg[g] = v;
    }
}

// one thread per (edge, 4 features): agg[dst] += xw[src] * coef[e]
__global__ void edge_scatter(const float* __restrict__ xw, float* __restrict__ agg,
                             const int* __restrict__ src, const int* __restrict__ dst,
                             const float* __restrict__ coef, int total /* E*32 */) {
    for (int g = blockIdx.x * blockDim.x + threadIdx.x; g < total; g += gridDim.x * blockDim.x) {
        int e = g >> 5, f4 = (g & 31) << 2;
        float c = coef[e];
        float4 v = *(const float4*)&xw[(size_t)src[e] * 128 + f4];
        float* d = &agg[(size_t)dst[e] * 128 + f4];
        unsafeAtomicAdd(d + 0, v.x * c);
        unsafeAtomicAdd(d + 1, v.y * c);
        unsafeAtomicAdd(d + 2, v.z * c);
        unsafeAtomicAdd(d + 3, v.w * c);
    }
}

// ---------------------------------------------------------------- pooling: out[g] += relu(agg2 + b2), batch sorted
__global__ __launch_bounds__(128) void pool_sorted(const float* __restrict__ agg,
                                                   const float* __restrict__ b2,
                                                   const int* __restrict__ batch,
                                                   float* __restrict__ out,
                                                   int N, int nodesPerBlock) {
    int f = threadIdx.x;
    int start = blockIdx.x * nodesPerBlock;
    int end = start + nodesPerBlock; if (end > N) end = N;
    if (start >= end) return;
    float bias = b2[f];
    float run = 0.0f;
    int gprev = batch[start];
    for (int i = start; i < end; ++i) {
        int g = batch[i];
        if (g != gprev) {
            unsafeAtomicAdd(&out[(size_t)gprev * 128 + f], run);
            run = 0.0f; gprev = g;
        }
        float v = agg[(size_t)i * 128 + f] + bias;
        run += v > 0.0f ? v : 0.0f;
    }
    unsafeAtomicAdd(&out[(size_t)gprev * 128 + f], run);
}

// ---------------------------------------------------------------- driver
extern "C" void kernel_launch(void* const* d_in, const int* in_sizes, int n_in,
                              void* d_out, int out_size, void* d_ws, size_t ws_size,
                              hipStream_t stream) {
    const float* x    = (const float*)d_in[0];
    const int*   ei   = (const int*)d_in[1];      // [2,E]: src then dst
    const int*   batch= (const int*)d_in[2];
    const float* W1   = (const float*)d_in[4];
    const float* b1   = (const float*)d_in[5];
    const float* W2   = (const float*)d_in[6];
    const float* b2   = (const float*)d_in[7];
    float* out = (float*)d_out;

    const int N = in_sizes[0] / 128;
    const int E = in_sizes[1] / 2;
    const int* src = ei;
    const int* dst = ei + E;

    // workspace layout (floats)
    float* ws       = (float*)d_ws;
    float* sums     = ws;             // 128
    float* sumsq    = ws + 128;       // 128
    float* inv_sig  = ws + 256;       // 128
    float* mu_os    = ws + 384;       // 128
    float* c1       = ws + 512;       // 128
    float* W1p      = ws + 640;       // 16384
    float* dinv     = ws + 17024;     // N  (degree, then rsqrt in place)
    float* coef     = dinv + N;       // E
    float* buf_xw   = coef + E;       // N*128  (xw1, then xw2)
    float* buf_agg  = buf_xw + (size_t)N * 128;  // N*128 (agg1, then agg2)

    const int totNF4 = N * 32;        // float4 elements
    const int totEF4 = E * 32;

    // 1) column stats -> folded standardization
    zero_f32<<<1, 256, 0, stream>>>(sums, 256);
    col_stats<<<1024, 128, 0, stream>>>(x, sums, sumsq, N);
    stats_finalize<<<1, 128, 0, stream>>>(sums, sumsq, inv_sig, mu_os, N);
    build_w1p<<<128, 128, 0, stream>>>(W1, inv_sig, W1p);
    build_c1<<<1, 128, 0, stream>>>(W1, mu_os, c1);

    // 2) degree -> dinv -> per-edge coef
    zero_f32<<<(N + 255) / 256, 256, 0, stream>>>(dinv, N);
    deg_count<<<(E + 255) / 256, 256, 0, stream>>>(dst, dinv, E);
    make_dinv<<<(N + 255) / 256, 256, 0, stream>>>(dinv, N);
    make_coef<<<(E + 255) / 256, 256, 0, stream>>>(src, dst, dinv, coef, E);

    int gtiles = (N + 15) / 16;
    int gblocks = gtiles < 4096 ? gtiles : 4096;

    // 3) layer 1: xw1 = x @ W1p + c1 ; agg1 = self + scatter
    gemm128_wmma<true><<<gblocks, 128, 0, stream>>>(x, W1p, nullptr, c1, buf_xw, N);
    self_init<<<(totNF4 + 255) / 256, 256, 0, stream>>>((const float4*)buf_xw, dinv, (float4*)buf_agg, totNF4);
    edge_scatter<<<(totEF4 + 255) / 256, 256, 0, stream>>>(buf_xw, buf_agg, src, dst, coef, totEF4);

    // 4) layer 2: xw2 = relu(agg1 + b1) @ W2 ; agg2 = self + scatter
    gemm128_wmma<false><<<gblocks, 128, 0, stream>>>(buf_agg, W2, b1, nullptr, buf_xw, N);
    self_init<<<(totNF4 + 255) / 256, 256, 0, stream>>>((const float4*)buf_xw, dinv, (float4*)buf_agg, totNF4);
    edge_scatter<<<(totEF4 + 255) / 256, 256, 0, stream>>>(buf_xw, buf_agg, src, dst, coef, totEF4);

    // 5) pool: out[g] = sum_i relu(agg2[i] + b2), batch sorted
    zero_f32<<<(out_size + 255) / 256, 256, 0, stream>>>(out, out_size);
    const int npb = 512;
    pool_sorted<<<(N + npb - 1) / npb, 128, 0, stream>>>(buf_agg, b2, batch, out, N, npb);
}